// CWRRTEWindowCell_63823214018890
// MI455X (gfx1250) — compile-verified
//
#include <hip/hip_runtime.h>
#include <math.h>

// Problem constants (match reference)
#define B_ 2048
#define K_ 64
#define D_ 512
#define DX_ 514            // D + 2 (concat write_strength, novelty)
#define EPSF 1e-6f

typedef float v2f __attribute__((ext_vector_type(2)));
typedef float v8f __attribute__((ext_vector_type(8)));

// ---------------------------------------------------------------------------
// Kernel 1: gate MLP  g = sigmoid( silu([q,ws,nov] @ W1 + b1) @ W2 + b2 )
// WMMA f32 16x16x4: one block = 16 rows of x; 8 waves x 4 N-tiles of 16 cols.
// Branch-free main K loop (k = 0..511), single tail WMMA for k = 512..515.
// ---------------------------------------------------------------------------
__global__ __launch_bounds__(256) void mlp_gate_kernel(
    const float* __restrict__ q_win,   // (B, D)
    const float* __restrict__ wstr,    // (B, 1)
    const float* __restrict__ nov,     // (B, 1)
    const float* __restrict__ W1,      // (D+2, D) row-major
    const float* __restrict__ b1,      // (D)
    const float* __restrict__ W2,      // (D, 1)
    const float* __restrict__ b2,      // (1)
    float* __restrict__ g_out)         // (B)
{
    __shared__ float lds_g[16];

    const int tid  = threadIdx.x;
    const int lane = tid & 31;
    const int wave = tid >> 5;
    const int half = lane >> 4;   // 0: lanes 0-15, 1: lanes 16-31
    const int lm   = lane & 15;
    const int rowBase = blockIdx.x * 16;
    const int row = rowBase + lm;       // A-matrix: M = lane%16 in both halves
    const int colBase = wave * 64 + lm; // this wave's 4 N-tiles: +0,+16,+32,+48

    if (tid < 16) lds_g[tid] = 0.0f;
    __syncthreads();

    v8f c0, c1, c2, c3;
    #pragma unroll
    for (int j = 0; j < 8; ++j) { c0[j] = 0.f; c1[j] = 0.f; c2[j] = 0.f; c3[j] = 0.f; }

    const float* qrow = q_win + (size_t)row * D_;

    // ---- main K loop: fully unconditional loads --------------------------
    for (int k = 0; k < D_; k += 4) {
        const int ka = k + 2 * half;                 // even, < 512
        const v2f a = *(const v2f*)&qrow[ka];        // A: {x[ka], x[ka+1]}
        const float* w1r0 = W1 + (size_t)ka * D_;    // W1 row ka
        const float* w1r1 = w1r0 + D_;               // W1 row ka+1
        v2f bb0, bb1, bb2, bb3;
        bb0.x = w1r0[colBase];      bb0.y = w1r1[colBase];
        bb1.x = w1r0[colBase + 16]; bb1.y = w1r1[colBase + 16];
        bb2.x = w1r0[colBase + 32]; bb2.y = w1r1[colBase + 32];
        bb3.x = w1r0[colBase + 48]; bb3.y = w1r1[colBase + 48];
        c0 = __builtin_amdgcn_wmma_f32_16x16x4_f32(false, a, false, bb0, (short)0, c0, false, false);
        c1 = __builtin_amdgcn_wmma_f32_16x16x4_f32(false, a, false, bb1, (short)0, c1, false, false);
        c2 = __builtin_amdgcn_wmma_f32_16x16x4_f32(false, a, false, bb2, (short)0, c2, false, false);
        c3 = __builtin_amdgcn_wmma_f32_16x16x4_f32(false, a, false, bb3, (short)0, c3, false, false);
    }

    // ---- tail: k = 512..515. half0 sees {ws, nov}; half1 sees K=514,515 -> 0.
    // OOB W1 rows (514,515) are clamped to row 513: their A entries are 0,
    // so the products contribute nothing.
    {
        v2f a;
        a.x = half ? 0.0f : wstr[row];   // v_cndmask, no branch
        a.y = half ? 0.0f : nov[row];
        const float* w1r0 = W1 + (size_t)(half ? 513 : 512) * D_;
        const float* w1r1 = W1 + (size_t)513 * D_;
        v2f bb0, bb1, bb2, bb3;
        bb0.x = w1r0[colBase];      bb0.y = w1r1[colBase];
        bb1.x = w1r0[colBase + 16]; bb1.y = w1r1[colBase + 16];
        bb2.x = w1r0[colBase + 32]; bb2.y = w1r1[colBase + 32];
        bb3.x = w1r0[colBase + 48]; bb3.y = w1r1[colBase + 48];
        c0 = __builtin_amdgcn_wmma_f32_16x16x4_f32(false, a, false, bb0, (short)0, c0, false, false);
        c1 = __builtin_amdgcn_wmma_f32_16x16x4_f32(false, a, false, bb1, (short)0, c1, false, false);
        c2 = __builtin_amdgcn_wmma_f32_16x16x4_f32(false, a, false, bb2, (short)0, c2, false, false);
        c3 = __builtin_amdgcn_wmma_f32_16x16x4_f32(false, a, false, bb3, (short)0, c3, false, false);
    }

    // ---- Epilogue: bias + SiLU + dot with W2, reduce into per-row gates ---
    const v8f cc[4] = { c0, c1, c2, c3 };
    #pragma unroll
    for (int t = 0; t < 4; ++t) {
        const int col = colBase + t * 16;
        const float w2v = W2[col];
        const float b1v = b1[col];
        #pragma unroll
        for (int j = 0; j < 8; ++j) {
            // C/D layout: element j on this lane is row (j + 8*half)
            float h  = cc[t][j] + b1v;
            float hv = h / (1.0f + __expf(-h));   // silu
            float contrib = hv * w2v;
            // reduce across the 16 lanes of this half (same row)
            for (int m = 1; m < 16; m <<= 1)
                contrib += __shfl_xor(contrib, m, 32);
            if (lm == 0) atomicAdd(&lds_g[j + 8 * half], contrib);
        }
    }
    __syncthreads();

    if (tid < 16) {
        float s = lds_g[tid] + b2[0];
        g_out[rowBase + tid] = 1.0f / (1.0f + __expf(-s));
    }
}

// ---------------------------------------------------------------------------
// Block-wide sum reduction (256 threads = 8 waves). scr must hold >= 9 floats.
// ---------------------------------------------------------------------------
__device__ __forceinline__ float block_reduce_sum(float v, float* scr) {
    for (int m = 16; m >= 1; m >>= 1) v += __shfl_xor(v, m, 32);
    __syncthreads();                       // protect scr from prior use
    if ((threadIdx.x & 31) == 0) scr[threadIdx.x >> 5] = v;
    __syncthreads();
    if (threadIdx.x == 0) {
        float s = 0.0f;
        for (int i = 0; i < 8; ++i) s += scr[i];
        scr[8] = s;
    }
    __syncthreads();
    return scr[8];
}

// ---------------------------------------------------------------------------
// Kernel 2: one workgroup (256 thr) per batch row. Streams keys once
// (3 dot reductions + copy-out), softmax + slot selection, streams vals once
// (weighted read + copy-out), then rewrites the single modified slot row.
// ---------------------------------------------------------------------------
__global__ __launch_bounds__(256) void epi_row_kernel(
    const float* __restrict__ q_win,      // (B,D)
    const float* __restrict__ write_key,  // (B,D)
    const float* __restrict__ write_val,  // (B,D)
    const float* __restrict__ wstr,       // (B,1)
    const float* __restrict__ keys,       // (B,K,D)
    const float* __restrict__ vals,       // (B,K,D)
    const float* __restrict__ age,        // (B,K)
    const float* __restrict__ strength,   // (B,K)
    const float* __restrict__ scale,      // (D)
    const float* __restrict__ g_arr,      // (B) from kernel 1
    float* __restrict__ out_read,         // (B,D)
    float* __restrict__ out_keys,         // (B,K,D)
    float* __restrict__ out_vals,         // (B,K,D)
    float* __restrict__ out_age,          // (B,K)
    float* __restrict__ out_str)          // (B,K)
{
    __shared__ __align__(16) float s_q[D_];
    __shared__ __align__(16) float s_wk[D_];
    __shared__ float s_dq[K_], s_dw[K_], s_nrm[K_], s_w[K_];
    __shared__ float scr[9];
    __shared__ float s_invq, s_invwk, s_upd, s_alpha, s_beta;
    __shared__ int   s_slot;

    const int b    = blockIdx.x;
    const int tid  = threadIdx.x;
    const int lane = tid & 31;
    const int wave = tid >> 5;

    const float* keys_row = keys + (size_t)b * K_ * D_;
    const float* vals_row = vals + (size_t)b * K_ * D_;
    float* okeys_row = out_keys + (size_t)b * K_ * D_;
    float* ovals_row = out_vals + (size_t)b * K_ * D_;

    // ---- Stage A: load q / write_key into LDS, compute their norms --------
    float2 qv = *(const float2*)&q_win[b * D_ + 2 * tid];
    float2 wv = *(const float2*)&write_key[b * D_ + 2 * tid];
    s_q[2 * tid]      = qv.x;  s_q[2 * tid + 1]  = qv.y;
    s_wk[2 * tid]     = wv.x;  s_wk[2 * tid + 1] = wv.y;

    float qn2 = block_reduce_sum(qv.x * qv.x + qv.y * qv.y, scr);
    float wn2 = block_reduce_sum(wv.x * wv.x + wv.y * wv.y, scr);
    if (tid == 0) {
        s_invq  = 1.0f / (sqrtf(qn2) + EPSF);
        s_invwk = 1.0f / (sqrtf(wn2) + EPSF);
    }
    __syncthreads();

    // ---- Stage B: stream keys: dots with q & wk, key norms, copy-out ------
    for (int kk = 0; kk < 8; ++kk) {
        const int k = wave * 8 + kk;
        const float* kp = keys_row + k * D_;
        // warm L2 for the vals pass (Stage D) while we stream keys
        __builtin_prefetch(vals_row + k * D_ + lane * 16, 0, 0);
        float aq = 0.0f, aw = 0.0f, an = 0.0f;
        #pragma unroll
        for (int cch = 0; cch < 4; ++cch) {
            const int d = cch * 128 + lane * 4;
            float4 kv = *(const float4*)&kp[d];
            *(float4*)&okeys_row[k * D_ + d] = kv;          // bulk copy-out
            float4 q4 = *(const float4*)&s_q[d];
            float4 w4 = *(const float4*)&s_wk[d];
            aq += kv.x * q4.x + kv.y * q4.y + kv.z * q4.z + kv.w * q4.w;
            aw += kv.x * w4.x + kv.y * w4.y + kv.z * w4.z + kv.w * w4.w;
            an += kv.x * kv.x + kv.y * kv.y + kv.z * kv.z + kv.w * kv.w;
        }
        for (int m = 16; m >= 1; m >>= 1) {
            aq += __shfl_xor(aq, m, 32);
            aw += __shfl_xor(aw, m, 32);
            an += __shfl_xor(an, m, 32);
        }
        if (lane == 0) { s_dq[k] = aq; s_dw[k] = aw; s_nrm[k] = sqrtf(an); }
    }
    __threadfence_block();   // order bulk key copies before later slot rewrite
    __syncthreads();

    // ---- Stage C: softmax weights + slot selection (serial, tiny) ---------
    if (tid == 0) {
        const float invq = s_invq, invw = s_invwk;
        const float* str = strength + (size_t)b * K_;
        const float* ag  = age + (size_t)b * K_;
        float mx = -1e30f;
        for (int k = 0; k < K_; ++k) {
            const float st   = str[k];
            const float invk = 1.0f / (s_nrm[k] + EPSF);
            const float sim  = s_dq[k] * invq * invk;
            const float scl  = fminf(fmaxf(st, 1e-3f), 1e9f);
            float lg = sim + 0.5f * __logf(scl) - 0.02f * ag[k];
            if (!(st > 1e-3f)) lg -= 1000.0f;
            s_w[k] = lg;                       // stash logits in s_w
            mx = fmaxf(mx, lg);
        }
        float ssum = 0.0f;
        for (int k = 0; k < K_; ++k) {
            const float e = __expf(s_w[k] - mx);
            s_w[k] = e; ssum += e;
        }
        const float inv_s = 1.0f / ssum;
        for (int k = 0; k < K_; ++k) s_w[k] *= inv_s;

        float bestsim = -1e30f, bestage = -1e30f;
        int bidx = 0, lidx = 0;
        for (int k = 0; k < K_; ++k) {
            const float st   = str[k];
            const float invk = 1.0f / (s_nrm[k] + EPSF);
            const float simw = s_dw[k] * invw * invk;
            if (simw > bestsim) { bestsim = simw; bidx = k; }
            const float ascore = ag[k] + (1.0f - fminf(fmaxf(st, 0.0f), 1.0f)) * 0.01f;
            if (ascore > bestage) { bestage = ascore; lidx = k; }
        }
        const int slot = (bestsim > 0.85f) ? bidx : lidx;
        const float ws = fminf(fmaxf(wstr[b], 0.0f), 1.0f);
        const float prev = str[slot] * 0.999f;
        s_slot  = slot;
        s_upd   = fminf(fmaxf(prev + ws * (1.0f - prev), 1e-3f), 1.0f);
        s_alpha = 0.25f * ws;
        s_beta  = 0.25f * ws;
    }
    __syncthreads();

    // ---- age / strength outputs ------------------------------------------
    if (tid < K_) {
        const int k = tid;
        out_age[(size_t)b * K_ + k] =
            (k == s_slot) ? 0.0f : (age[(size_t)b * K_ + k] + 1.0f);
        out_str[(size_t)b * K_ + k] =
            (k == s_slot) ? s_upd : (strength[(size_t)b * K_ + k] * 0.999f);
    }

    // ---- Stage D: stream vals: weighted read + copy-out -------------------
    const int d0 = 2 * tid;
    float accx = 0.0f, accy = 0.0f;
    #pragma unroll 4
    for (int k = 0; k < K_; ++k) {
        float2 v = *(const float2*)&vals_row[k * D_ + d0];
        *(float2*)&ovals_row[k * D_ + d0] = v;              // bulk copy-out
        const float wk = s_w[k];
        accx += wk * v.x;
        accy += wk * v.y;
    }
    __threadfence_block();   // order bulk val copies before slot rewrite

    // ---- Stage E: rmsnorm + gate, then rewrite the modified slot row ------
    const float ss  = block_reduce_sum(accx * accx + accy * accy, scr);
    const float inv_rms = 1.0f / sqrtf(ss / (float)D_ + EPSF);
    const float gb  = g_arr[b];
    out_read[(size_t)b * D_ + d0]     = gb * (accx * inv_rms * scale[d0]);
    out_read[(size_t)b * D_ + d0 + 1] = gb * (accy * inv_rms * scale[d0 + 1]);

    const int   slot  = s_slot;
    const float alpha = s_alpha, beta = s_beta;

    // blended, re-normalized key at slot
    float2 okv = *(const float2*)&keys_row[slot * D_ + d0];
    const float nkx = (1.0f - alpha) * okv.x + alpha * s_wk[d0];
    const float nky = (1.0f - alpha) * okv.y + alpha * s_wk[d0 + 1];
    const float nsum = block_reduce_sum(nkx * nkx + nky * nky, scr);
    const float invn = 1.0f / (sqrtf(nsum) + EPSF);
    *(float2*)&okeys_row[slot * D_ + d0] = make_float2(nkx * invn, nky * invn);

    // blended val at slot
    float2 ovv = *(const float2*)&vals_row[slot * D_ + d0];
    float2 wvv = *(const float2*)&write_val[(size_t)b * D_ + d0];
    *(float2*)&ovals_row[slot * D_ + d0] =
        make_float2((1.0f - beta) * ovv.x + beta * wvv.x,
                    (1.0f - beta) * ovv.y + beta * wvv.y);
}

// ---------------------------------------------------------------------------
extern "C" void kernel_launch(void* const* d_in, const int* in_sizes, int n_in,
                              void* d_out, int out_size, void* d_ws, size_t ws_size,
                              hipStream_t stream) {
    const float* q_win   = (const float*)d_in[0];
    const float* wkey    = (const float*)d_in[1];
    const float* wval    = (const float*)d_in[2];
    const float* wstr    = (const float*)d_in[3];
    const float* nov     = (const float*)d_in[4];
    const float* keys    = (const float*)d_in[5];
    const float* vals    = (const float*)d_in[6];
    const float* age     = (const float*)d_in[7];
    const float* strg    = (const float*)d_in[8];
    const float* scale   = (const float*)d_in[9];
    const float* W1      = (const float*)d_in[10];
    const float* b1      = (const float*)d_in[11];
    const float* W2      = (const float*)d_in[12];
    const float* b2      = (const float*)d_in[13];

    float* out_read = (float*)d_out;
    float* out_keys = out_read + (size_t)B_ * D_;
    float* out_vals = out_keys + (size_t)B_ * K_ * D_;
    float* out_age  = out_vals + (size_t)B_ * K_ * D_;
    float* out_str  = out_age  + (size_t)B_ * K_;

    float* g_ws = (float*)d_ws;   // B floats of scratch for the gate

    mlp_gate_kernel<<<B_ / 16, 256, 0, stream>>>(
        q_win, wstr, nov, W1, b1, W2, b2, g_ws);

    epi_row_kernel<<<B_, 256, 0, stream>>>(
        q_win, wkey, wval, wstr, keys, vals, age, strg, scale, g_ws,
        out_read, out_keys, out_vals, out_age, out_str);
}